// Net2_48481590837926
// MI455X (gfx1250) — compile-verified
//
#include <hip/hip_runtime.h>
#include <hip/hip_bf16.h>
#include <math.h>

// Problem constants (match reference)
#define DDIM   100   // state dimension
#define NSTEPS 40    // SDE steps
#define NPAD   48    // 40 padded to 3 WMMA M-tiles of 16

typedef __attribute__((ext_vector_type(2))) float v2f;
typedef __attribute__((ext_vector_type(8))) float v8f;

// Faithful translation of the drift with the 3-way data-dependent branch.
__device__ __forceinline__ float drift_f(float u) {
    float low  = -0.02f  * u / 3.0f - 0.02f * u;                       // u < 50
    float high = -0.002f * u / 3.0f - 0.02f * u;                       // u >= 70
    float mid  = (-(70.0f - 50.0f) / (0.02f - 0.2f)) * (u - 50.0f) / 3.0f * u
                 - (0.2f / 3.0f) * u - 0.02f * u;                      // 50 <= u < 70
    float r = (u >= 70.0f) ? high : mid;
    return (u < 50.0f) ? low : r;
}

// One block, 4 waves (wave32). Computes u_f only — the per-step MLPs in the
// reference never influence the returned value, so they are skipped entirely.
__global__ __launch_bounds__(128)
void bsde_u_kernel(const float* __restrict__ x0,
                   const float* __restrict__ tlist,
                   const float* __restrict__ noise,   // (NSTEPS, DDIM, 1) flat
                   const float* __restrict__ u0,
                   const float* __restrict__ gu0,     // (DDIM, 1) flat
                   float* __restrict__ out) {
    __shared__ float sGu[DDIM + 4];   // gu = 0.2 * x0 * gu0, +pad so kk+bk+1 reads stay in-bounds
    __shared__ float sDot[NPAD];      // raw dot(gu, noise[k,:]) per step

    const int tid = threadIdx.x;

    if (tid < DDIM + 4) sGu[tid] = 0.0f;
    __syncthreads();
    if (tid < DDIM) sGu[tid] = 0.2f * x0[tid] * gu0[tid];
    __syncthreads();

    // ---- Phase 2: 40x100 GEMV via V_WMMA_F32_16X16X4_F32 on wave 0 ----
    // A: 16 steps x 4 K-elems of noise; B: gu[k] replicated across 16 columns.
    // Branch is uniform per wave -> EXEC all-ones inside, as WMMA requires.
    if (tid < 32) {
        const int lane = tid;
        const int m    = lane & 15;              // row within M-tile
        const int bk   = (lane < 16) ? 0 : 2;    // K sub-offset per ISA A-layout
        for (int t = 0; t < 3; ++t) {
            const int   step  = t * 16 + m;
            const float mask  = (step < NSTEPS) ? 1.0f : 0.0f;  // pad rows -> 0 (no divergence)
            const int   cstep = (step < NSTEPS) ? step : 0;     // clamp address, keep loads converged
            const float* row  = noise + cstep * DDIM;
            v8f c = {0.f, 0.f, 0.f, 0.f, 0.f, 0.f, 0.f, 0.f};
            #pragma unroll
            for (int kk = 0; kk < DDIM; kk += 4) {   // 100 = 25 exact chunks of K=4
                v2f a, b;
                a.x = row[kk + bk]     * mask;
                a.y = row[kk + bk + 1] * mask;
                b.x = sGu[kk + bk];          // same value for every column n
                b.y = sGu[kk + bk + 1];
                c = __builtin_amdgcn_wmma_f32_16x16x4_f32(
                        /*neg_a=*/false, a, /*neg_b=*/false, b,
                        /*c_mod=*/(short)0, c, /*reuse_a=*/false, /*reuse_b=*/false);
            }
            // D layout: lane (n + 16*(mhi)) holds M = v + (lane<16 ? 0 : 8) in VGPR v.
            // Column n=0 lives in lanes 0 and 16.
            if (m == 0) {
                const int mbase = (lane < 16) ? 0 : 8;
                #pragma unroll
                for (int v = 0; v < 8; ++v)
                    sDot[t * 16 + mbase + v] = c[v];
            }
        }
    }
    __syncthreads();

    // ---- Phase 3: inherently sequential scalar recurrence (40 steps) ----
    if (tid == 0) {
        float u = u0[0];
        #pragma unroll 4
        for (int k = 0; k < NSTEPS; ++k) {
            const float dt    = tlist[k];
            const float term3 = sDot[k] * sqrtf(dt);   // dot(gu, dW), dW = noise*sqrt(dt)
            u = u - drift_f(u) * dt + term3;
        }
        out[0] = u;
    }
}

extern "C" void kernel_launch(void* const* d_in, const int* in_sizes, int n_in,
                              void* d_out, int out_size, void* d_ws, size_t ws_size,
                              hipStream_t stream) {
    (void)in_sizes; (void)n_in; (void)out_size; (void)d_ws; (void)ws_size;
    const float* x0    = (const float*)d_in[0];
    const float* tlist = (const float*)d_in[1];
    const float* noise = (const float*)d_in[2];
    const float* u0    = (const float*)d_in[3];
    const float* gu0   = (const float*)d_in[4];
    // d_in[5..10] (W1,b1,W2,b2,W3,b3) are dead w.r.t. the returned u_f: skipped.
    float* out = (float*)d_out;
    bsde_u_kernel<<<1, 128, 0, stream>>>(x0, tlist, noise, u0, gu0, out);
}